// GraphUNet_42786464203097
// MI455X (gfx1250) — compile-verified
//
#include <hip/hip_runtime.h>
#include <math.h>

// ---------------------------------------------------------------------------
// Graph U-Net (dense) for MI455X / gfx1250.
// All GEMMs run on V_WMMA_F32_16X16X4_F32 (exact fp32 matrix pipe, wave32).
// GEMM: 128x64 block tile, 32x32 per wave (4 accumulators), double-buffered
// LDS staged with b128 loads/stores, bank-conflict-free padded strides.
// ---------------------------------------------------------------------------

typedef __attribute__((ext_vector_type(2))) float v2f;
typedef __attribute__((ext_vector_type(8))) float v8f;

#define GN 4096
#define GHID 256
#define GINC 128
#define GOUTC 128

static inline int cdiv(long a, long b) { return (int)((a + b - 1) / b); }

// Padded LDS row strides (floats). Both multiples of 4 -> b128-aligned rows.
// SA=20: a-frag rows r*20 mod 64 all distinct over 16 rows; +2 col offset of
// lanes 16-31 lands on banks ==2 mod 4, disjoint from ==0 mod 4 row banks.
// SB=72: lanes 16-31 read row +2 -> bank offset +16, disjoint from lanes 0-15.
#define SA 20
#define SB 72

__global__ __launch_bounds__(256) void wmma_gemm_f32(
    float* __restrict__ C, const float* __restrict__ A, const float* __restrict__ B,
    int M, int N, int K) {
  __shared__ float As[2][128 * SA];  // 128 rows x 16 k per stage
  __shared__ float Bs[2][16 * SB];   // 16 k x 64 cols per stage

  const int tid  = threadIdx.x;
  const int wave = tid >> 5;
  const int lane = tid & 31;
  const int row0 = blockIdx.y * 128;
  const int col0 = blockIdx.x * 64;
  const int wr   = (wave & 3) * 32;   // wave row offset within block tile
  const int wc   = (wave >> 2) * 32;  // wave col offset within block tile
  const int am   = lane & 15;         // M (or N) index within fragment
  const int ak   = (lane >> 4) << 1;  // lanes 16..31 hold K=2,3 of each step

  // Staging map: A tile 128x16 = 512 float4 (2/thread), B tile 16x64 = 256 (1/thread)
  const int a_r = tid >> 2;          // 0..63 (second half at +64)
  const int a_c = (tid & 3) * 4;
  const int b_r = tid >> 4;          // 0..15
  const int b_c = (tid & 15) * 4;

  v8f acc00 = {}, acc01 = {}, acc10 = {}, acc11 = {};
  float4 ra0, ra1, rb;

  // fetch global -> regs for K-step k0
  auto fetch = [&](int k0) {
    ra0 = *(const float4*)&A[(size_t)(row0 + a_r) * K + k0 + a_c];
    ra1 = *(const float4*)&A[(size_t)(row0 + a_r + 64) * K + k0 + a_c];
    rb  = *(const float4*)&B[(size_t)(k0 + b_r) * N + col0 + b_c];
  };
  // regs -> LDS stage s
  auto stage = [&](int s) {
    *(float4*)&As[s][a_r * SA + a_c]        = ra0;
    *(float4*)&As[s][(a_r + 64) * SA + a_c] = ra1;
    *(float4*)&Bs[s][b_r * SB + b_c]        = rb;
  };

  fetch(0);
  stage(0);
  __syncthreads();

  int s = 0;
  for (int k0 = 0; k0 < K; k0 += 16) {
    const bool more = (k0 + 16) < K;
    if (more) {
      fetch(k0 + 16);
      if (k0 + 32 < K) {
        __builtin_prefetch(&A[(size_t)(row0 + a_r) * K + k0 + 32], 0, 1);
        __builtin_prefetch(&B[(size_t)(k0 + 32 + b_r) * N + col0 + b_c], 0, 1);
      }
    }

#pragma unroll
    for (int kk = 0; kk < 16; kk += 4) {
      v2f a0, a1, b0, b1;
      // A fragment 16x4: lanes 0-15 -> K=kk,kk+1; lanes 16-31 -> K=kk+2,kk+3
      a0.x = As[s][(wr + am) * SA + kk + ak];
      a0.y = As[s][(wr + am) * SA + kk + ak + 1];
      a1.x = As[s][(wr + 16 + am) * SA + kk + ak];
      a1.y = As[s][(wr + 16 + am) * SA + kk + ak + 1];
      // B fragment 4x16: K rows striped across VGPRs, N across lanes
      b0.x = Bs[s][(kk + ak) * SB + wc + am];
      b0.y = Bs[s][(kk + ak + 1) * SB + wc + am];
      b1.x = Bs[s][(kk + ak) * SB + wc + 16 + am];
      b1.y = Bs[s][(kk + ak + 1) * SB + wc + 16 + am];
      acc00 = __builtin_amdgcn_wmma_f32_16x16x4_f32(false, a0, false, b0,
                                                    (short)0, acc00, false, false);
      acc01 = __builtin_amdgcn_wmma_f32_16x16x4_f32(false, a0, false, b1,
                                                    (short)0, acc01, false, false);
      acc10 = __builtin_amdgcn_wmma_f32_16x16x4_f32(false, a1, false, b0,
                                                    (short)0, acc10, false, false);
      acc11 = __builtin_amdgcn_wmma_f32_16x16x4_f32(false, a1, false, b1,
                                                    (short)0, acc11, false, false);
    }

    if (more) stage(s ^ 1);
    __syncthreads();
    s ^= 1;
  }

  // C/D layout: VGPR r -> M = r (lanes 0-15) or 8+r (lanes 16-31), N = lane&15
  const int crow = row0 + wr + ((lane >> 4) << 3);
  const int ccol = col0 + wc + am;
#pragma unroll
  for (int r = 0; r < 8; ++r) {
    C[(size_t)(crow + r) * N + ccol]           = acc00[r];
    C[(size_t)(crow + r) * N + ccol + 16]      = acc01[r];
    C[(size_t)(crow + 16 + r) * N + ccol]      = acc10[r];
    C[(size_t)(crow + 16 + r) * N + ccol + 16] = acc11[r];
  }
}

// ---------------------------------------------------------------------------
// Elementwise / graph kernels
// ---------------------------------------------------------------------------

__global__ void fill_f32(float* __restrict__ dst, float v, long count) {
  long i = (long)blockIdx.x * blockDim.x + threadIdx.x;
  long stride = (long)gridDim.x * blockDim.x;
  for (; i < count; i += stride) dst[i] = v;
}

// A[u,v] = A[v,u] = 1 for every edge; self-loops skipped (diag stays 0).
__global__ void scatter_edges(float* __restrict__ A, const int* __restrict__ ei,
                              int E, int n) {
  int i = blockIdx.x * blockDim.x + threadIdx.x;
  if (i >= E) return;
  int u = ei[i];
  int v = ei[E + i];
  if (u != v) {
    A[(size_t)u * n + v] = 1.0f;
    A[(size_t)v * n + u] = 1.0f;
  }
}

// augment: T = A@A (already in T) + 2A, diagonal forced to 0.
// (A+I)@(A+I) = A@A + 2A + I, then remove self loops.
__global__ void aug_finish(float* __restrict__ T, const float* __restrict__ A, int n) {
  long i = (long)blockIdx.x * blockDim.x + threadIdx.x;
  long total = (long)n * n;
  if (i >= total) return;
  int r = (int)(i / n);
  int c = (int)(i % n);
  T[i] = (r == c) ? 0.0f : (T[i] + 2.0f * A[i]);
}

// dinv[r] = (rowsum(A) + 2)^-1/2   (Ah = A + 2I, diag(A)=0)
__global__ void dinv_kernel(float* __restrict__ dinv, const float* __restrict__ A, int n) {
  __shared__ float red[256];
  int row = blockIdx.x, t = threadIdx.x;
  const float* ar = A + (size_t)row * n;
  float s = 0.0f;
  for (int j = t; j < n; j += 256) s += ar[j];
  red[t] = s;
  __syncthreads();
  for (int k = 128; k > 0; k >>= 1) {
    if (t < k) red[t] += red[t + k];
    __syncthreads();
  }
  if (t == 0) dinv[row] = rsqrtf(red[0] + 2.0f);
}

__global__ void scale_rows(float* __restrict__ V, const float* __restrict__ Y,
                           const float* __restrict__ dinv, int n, int c) {
  long i = (long)blockIdx.x * blockDim.x + threadIdx.x;
  if (i >= (long)n * c) return;
  V[i] = Y[i] * dinv[i / c];
}

// out = dinv_r * Z + 2*dinv_r^2 * Y + b[col]   (== An @ (X@W) + b), opt ReLU
__global__ void gcn_finish(float* __restrict__ out, const float* __restrict__ Z,
                           const float* __restrict__ Y, const float* __restrict__ dinv,
                           const float* __restrict__ b, int n, int c, int relu) {
  long i = (long)blockIdx.x * blockDim.x + threadIdx.x;
  if (i >= (long)n * c) return;
  int r   = (int)(i / c);
  int col = (int)(i % c);
  float di = dinv[r];
  float v  = di * Z[i] + 2.0f * di * di * Y[i] + b[col];
  out[i] = relu ? fmaxf(v, 0.0f) : v;
}

__global__ void pnorm_kernel(float* __restrict__ pn, const float* __restrict__ p, int c) {
  __shared__ float red[256];
  int t = threadIdx.x;
  float v = (t < c) ? p[t] * p[t] : 0.0f;
  red[t] = v;
  __syncthreads();
  for (int k = 128; k > 0; k >>= 1) {
    if (t < k) red[t] += red[t + k];
    __syncthreads();
  }
  if (t == 0) pn[0] = sqrtf(red[0]);
}

__global__ void scores_kernel(float* __restrict__ score, const float* __restrict__ x,
                              const float* __restrict__ p, const float* __restrict__ pn,
                              int c) {
  __shared__ float red[256];
  int row = blockIdx.x, t = threadIdx.x;
  float v = (t < c) ? x[(size_t)row * c + t] * p[t] : 0.0f;
  red[t] = v;
  __syncthreads();
  for (int k = 128; k > 0; k >>= 1) {
    if (t < k) red[t] += red[t + k];
    __syncthreads();
  }
  if (t == 0) score[row] = tanhf(red[0] / pn[0]);
}

// Single-workgroup bitonic sort of (score, idx), descending score with
// ascending-index tie-break (matches jax.lax.top_k ordering). n pow2 <= 4096.
__global__ void bitonic_topk(int* __restrict__ perm, float* __restrict__ vals,
                             const float* __restrict__ score, int n, int k) {
  __shared__ float s[4096];
  __shared__ int   id[4096];
  int tid = threadIdx.x;
  for (int t = tid; t < n; t += blockDim.x) { s[t] = score[t]; id[t] = t; }
  __syncthreads();
  for (int sz = 2; sz <= n; sz <<= 1) {
    for (int j = sz >> 1; j > 0; j >>= 1) {
      for (int t = tid; t < n; t += blockDim.x) {
        int ixj = t ^ j;
        if (ixj > t) {
          float sa = s[t], sb = s[ixj];
          int   ia = id[t], ib = id[ixj];
          bool up     = ((t & sz) == 0);
          bool aFirst = (sa > sb) || (sa == sb && ia < ib);
          if (up != aFirst) {
            s[t] = sb; s[ixj] = sa;
            id[t] = ib; id[ixj] = ia;
          }
        }
      }
      __syncthreads();
    }
  }
  for (int t = tid; t < k; t += blockDim.x) { perm[t] = id[t]; vals[t] = s[t]; }
}

__global__ void gather_x(float* __restrict__ xn, const float* __restrict__ x,
                         const int* __restrict__ perm, const float* __restrict__ vals,
                         int k, int c) {
  long i = (long)blockIdx.x * blockDim.x + threadIdx.x;
  if (i >= (long)k * c) return;
  int r = (int)(i / c);
  xn[i] = x[(size_t)perm[r] * c + (i % c)] * vals[r];
}

__global__ void gather_A(float* __restrict__ An, const float* __restrict__ Abig,
                         const int* __restrict__ perm, int k, int nbig) {
  long i = (long)blockIdx.x * blockDim.x + threadIdx.x;
  if (i >= (long)k * k) return;
  int r = (int)(i / k);
  int c = (int)(i % k);
  An[i] = Abig[(size_t)perm[r] * nbig + perm[c]];
}

__global__ void copy_f32(float* __restrict__ dst, const float* __restrict__ src, long count) {
  long i = (long)blockIdx.x * blockDim.x + threadIdx.x;
  if (i < count) dst[i] = src[i];
}

// out[perm[r], :] += x[r, :]   (perm rows unique -> race-free plain add)
__global__ void scatter_add_rows(float* __restrict__ out, const float* __restrict__ x,
                                 const int* __restrict__ perm, int k, int c) {
  long i = (long)blockIdx.x * blockDim.x + threadIdx.x;
  if (i >= (long)k * c) return;
  int r = (int)(i / c);
  out[(size_t)perm[r] * c + (i % c)] += x[i];
}

// ---------------------------------------------------------------------------
// Host-side sequencing helpers (all on `stream`)
// ---------------------------------------------------------------------------

static void gemm(hipStream_t s, float* C, const float* A, const float* B,
                 int M, int N, int K) {
  dim3 grid(N / 64, M / 128);
  wmma_gemm_f32<<<grid, 256, 0, s>>>(C, A, B, M, N, K);
}

static void gcn(hipStream_t s, float* out, const float* x, const float* A,
                int n, int cin, int cout, const float* W, const float* b, int relu,
                float* Y, float* V, float* Z, float* dinv) {
  gemm(s, Y, x, W, n, cout, cin);                           // Y = X @ W
  dinv_kernel<<<n, 256, 0, s>>>(dinv, A, n);                // deg^-1/2
  long nc = (long)n * cout;
  scale_rows<<<cdiv(nc, 256), 256, 0, s>>>(V, Y, dinv, n, cout);
  gemm(s, Z, A, V, n, cout, n);                             // Z = A @ (dinv .* Y)
  gcn_finish<<<cdiv(nc, 256), 256, 0, s>>>(out, Z, Y, dinv, b, n, cout, relu);
}

static void augment(hipStream_t s, float* T, const float* A, int n) {
  gemm(s, T, A, A, n, n, n);                                // T = A @ A
  long nn = (long)n * n;
  aug_finish<<<cdiv(nn, 256), 256, 0, s>>>(T, A, n);        // + 2A, zero diag
}

static void pool(hipStream_t s, int* perm, float* vals, float* xn, float* An,
                 const float* x, const float* Abig, const float* p,
                 int n, int k, float* score, float* pn) {
  pnorm_kernel<<<1, 256, 0, s>>>(pn, p, GHID);
  scores_kernel<<<n, 256, 0, s>>>(score, x, p, pn, GHID);
  bitonic_topk<<<1, 1024, 0, s>>>(perm, vals, score, n, k);
  gather_x<<<cdiv((long)k * GHID, 256), 256, 0, s>>>(xn, x, perm, vals, k, GHID);
  gather_A<<<cdiv((long)k * k, 256), 256, 0, s>>>(An, Abig, perm, k, n);
}

// ---------------------------------------------------------------------------

extern "C" void kernel_launch(void* const* d_in, const int* in_sizes, int n_in,
                              void* d_out, int out_size, void* d_ws, size_t ws_size,
                              hipStream_t stream) {
  const float* x_in = (const float*)d_in[0];
  const int*   ei   = (const int*)d_in[1];
  const int    E    = in_sizes[1] / 2;
  const float* Wd0 = (const float*)d_in[2],  *bd0 = (const float*)d_in[3];
  const float* Wd1 = (const float*)d_in[4],  *bd1 = (const float*)d_in[5];
  const float* Wd2 = (const float*)d_in[6],  *bd2 = (const float*)d_in[7];
  const float* Wd3 = (const float*)d_in[8],  *bd3 = (const float*)d_in[9];
  const float* p1  = (const float*)d_in[10], *p2  = (const float*)d_in[11];
  const float* p3  = (const float*)d_in[12];
  const float* Wu0 = (const float*)d_in[13], *bu0 = (const float*)d_in[14];
  const float* Wu1 = (const float*)d_in[15], *bu1 = (const float*)d_in[16];
  const float* Wu2 = (const float*)d_in[17], *bu2 = (const float*)d_in[18];

  // Bump allocator over workspace (256B aligned slices).
  char*  base = (char*)d_ws;
  size_t off  = 0;
  auto alloc = [&](long floats) -> float* {
    float* p = (float*)(base + off);
    off += ((size_t)floats * 4 + 255) & ~(size_t)255;
    return p;
  };

  float* A0   = alloc((long)GN * GN);        // adjacency level 0 (kept)
  float* Tbig = alloc((long)GN * GN);        // augment scratch (reused)
  float* A1   = alloc(2048L * 2048);         // pooled adjacency level 1
  float* A2l  = alloc(1024L * 1024);         // pooled adjacency level 2
  float* A3   = alloc(512L * 512);           // pooled adjacency level 3
  float* x0   = alloc((long)GN * GHID);      // xs[0]
  float* x1   = alloc(2048L * GHID);         // xs[1]
  float* x2   = alloc(1024L * GHID);         // xs[2]
  float* xa   = alloc((long)GN * GHID);      // ping
  float* xb   = alloc((long)GN * GHID);      // pong
  float* Yb   = alloc((long)GN * GHID);      // gcn temp X@W
  float* Vb   = alloc((long)GN * GHID);      // gcn temp dinv.*Y
  float* Zb   = alloc((long)GN * GHID);      // gcn temp A@V
  float* dinv = alloc(GN);
  float* scr  = alloc(GN);
  float* vals = alloc(GN);
  float* pn   = alloc(64);
  int* perm1  = (int*)alloc(2048);
  int* perm2  = (int*)alloc(1024);
  int* perm3  = (int*)alloc(512);

  // ---- build adjacency ----------------------------------------------------
  fill_f32<<<4096, 256, 0, stream>>>(A0, 0.0f, (long)GN * GN);
  scatter_edges<<<cdiv(E, 256), 256, 0, stream>>>(A0, ei, E, GN);

  // ---- encoder level 0 ----------------------------------------------------
  gcn(stream, x0, x_in, A0, GN, GINC, GHID, Wd0, bd0, 1, Yb, Vb, Zb, dinv);

  // ---- down 1: 4096 -> 2048 ----------------------------------------------
  augment(stream, Tbig, A0, GN);
  pool(stream, perm1, vals, xa, A1, x0, Tbig, p1, GN, 2048, scr, pn);
  gcn(stream, x1, xa, A1, 2048, GHID, GHID, Wd1, bd1, 1, Yb, Vb, Zb, dinv);

  // ---- down 2: 2048 -> 1024 ----------------------------------------------
  augment(stream, Tbig, A1, 2048);
  pool(stream, perm2, vals, xa, A2l, x1, Tbig, p2, 2048, 1024, scr, pn);
  gcn(stream, x2, xa, A2l, 1024, GHID, GHID, Wd2, bd2, 1, Yb, Vb, Zb, dinv);

  // ---- down 3: 1024 -> 512 ------------------------------------------------
  augment(stream, Tbig, A2l, 1024);
  pool(stream, perm3, vals, xa, A3, x2, Tbig, p3, 1024, 512, scr, pn);
  gcn(stream, xb, xa, A3, 512, GHID, GHID, Wd3, bd3, 1, Yb, Vb, Zb, dinv);

  // ---- up 1: unpool to 1024, + xs[2], gcn(Wu0) relu ----------------------
  copy_f32<<<cdiv(1024L * GHID, 256), 256, 0, stream>>>(xa, x2, 1024L * GHID);
  scatter_add_rows<<<cdiv(512L * GHID, 256), 256, 0, stream>>>(xa, xb, perm3, 512, GHID);
  gcn(stream, xb, xa, A2l, 1024, GHID, GHID, Wu0, bu0, 1, Yb, Vb, Zb, dinv);

  // ---- up 2: unpool to 2048, + xs[1], gcn(Wu1) relu ----------------------
  copy_f32<<<cdiv(2048L * GHID, 256), 256, 0, stream>>>(xa, x1, 2048L * GHID);
  scatter_add_rows<<<cdiv(1024L * GHID, 256), 256, 0, stream>>>(xa, xb, perm2, 1024, GHID);
  gcn(stream, xb, xa, A1, 2048, GHID, GHID, Wu1, bu1, 1, Yb, Vb, Zb, dinv);

  // ---- up 3: unpool to 4096, + xs[0], gcn(Wu2) no relu -> output ---------
  copy_f32<<<cdiv((long)GN * GHID, 256), 256, 0, stream>>>(xa, x0, (long)GN * GHID);
  scatter_add_rows<<<cdiv(2048L * GHID, 256), 256, 0, stream>>>(xa, xb, perm1, 2048, GHID);
  gcn(stream, (float*)d_out, xa, A0, GN, GHID, GOUTC, Wu2, bu2, 0, Yb, Vb, Zb, dinv);
}